// TimeAggNet_64183991272028
// MI455X (gfx1250) — compile-verified
//
#include <hip/hip_runtime.h>
#include <hip/hip_bf16.h>
#include <math.h>

typedef __attribute__((ext_vector_type(2))) float v2f;
typedef __attribute__((ext_vector_type(8))) float v8f;

// sizes
#define BS 256
#define NN 128
#define BN (BS*NN)          // 32768
#define T_IN 1216
#define C1 32
#define L1 30               // pooled length after conv1+pool5
#define C2 64
#define L2C 10              // conv2 output length
#define C3 128
#define K2 96               // 32*3
#define K3 256              // 64*4

// ---------------------------------------------------------------------------
// Stage A: conv1 (Cin=1,K=10,stride=8,pad=1) + relu + maxpool(5,5) + relu
// out: [bn][j][c] (position-major, channel-fastest) so that conv2's im2col
// patch for position p is the contiguous 96-float run at 96*(10*bn+p).
// ---------------------------------------------------------------------------
__global__ void __launch_bounds__(256) conv1_pool1_kernel(
    const float* __restrict__ x, const float* __restrict__ w1,
    const float* __restrict__ b1, float* __restrict__ out) {
  __shared__ float sx[T_IN];
  __shared__ float sw[C1 * 10];
  __shared__ float sb[C1];
  __shared__ float so[C1 * L1];
  const int bn = blockIdx.x;
  const float* xrow = x + (size_t)bn * T_IN;
  for (int i = threadIdx.x; i < T_IN; i += 256) sx[i] = xrow[i];
  for (int i = threadIdx.x; i < C1 * 10; i += 256) sw[i] = w1[i];
  if (threadIdx.x < C1) sb[threadIdx.x] = b1[threadIdx.x];
  __syncthreads();
  for (int t = threadIdx.x; t < C1 * L1; t += 256) {
    const int c = t & 31;   // channel fastest across lanes (sx reads broadcast)
    const int j = t >> 5;   // pooled position 0..29
    float wr[10];
#pragma unroll
    for (int k = 0; k < 10; ++k) wr[k] = sw[c * 10 + k];
    const float bias = sb[c];
    float m = 0.0f;  // relu(max(relu(.))) == max(0, max(conv))
#pragma unroll
    for (int i = 0; i < 5; ++i) {
      const int p = 5 * j + i;          // conv position, 0..149
      const int base = 8 * p - 1;       // x index of tap k=0 (pad=1)
      float acc = bias;
#pragma unroll
      for (int k = 0; k < 10; ++k) {
        const int idx = base + k;       // < 1216 always; only p==0 hits -1
        const float xv = (idx >= 0) ? sx[idx] : 0.0f;
        acc = fmaf(xv, wr[k], acc);
      }
      m = fmaxf(m, acc);
    }
    so[j * C1 + c] = m;                  // [j][c]: stride-1 across lanes
  }
  __syncthreads();
  float* orow = out + (size_t)bn * (C1 * L1);
  for (int t = threadIdx.x; t < C1 * L1; t += 256) orow[t] = so[t];
}

// ---------------------------------------------------------------------------
// One-off: transpose w2 [co][ci][k] -> w2t [co][k*32+ci] to match the
// kidx' = k*32+ci ordering of the contiguous A patches.
// ---------------------------------------------------------------------------
__global__ void __launch_bounds__(256) w2t_kernel(
    const float* __restrict__ w2, float* __restrict__ w2t) {
  const int t = blockIdx.x * 256 + threadIdx.x;   // C2*K2 = 6144
  if (t >= C2 * K2) return;
  const int co = t / K2;
  const int kk = t % K2;        // target kidx' = k*32+ci
  const int k  = kk >> 5;
  const int ci = kk & 31;
  w2t[t] = w2[co * K2 + ci * 3 + k];
}

// ---------------------------------------------------------------------------
// Stage B: conv2 as GEMM via V_WMMA_F32_16X16X4_F32.
// A: stageA [M=BN*10][96] (row-major, contiguous), B: w2t [co][96]
// out: relu(D + b2) -> [M][64]
// ---------------------------------------------------------------------------
__global__ void __launch_bounds__(128) conv2_wmma_kernel(
    const float* __restrict__ A, const float* __restrict__ w2t,
    const float* __restrict__ b2, float* __restrict__ out) {
  const int lane  = threadIdx.x & 31;
  const int wave  = threadIdx.x >> 5;
  const int mtile = blockIdx.x * 4 + wave;     // 20480 m-tiles
  const int row   = lane & 15;                 // A row / B,D col
  const int khalf = lane >> 4;

  const float* arow = A + (size_t)(mtile * 16 + row) * K2 + khalf * 2;
  const float* wb0 = w2t + (size_t)(0  + row) * K2 + khalf * 2;
  const float* wb1 = w2t + (size_t)(16 + row) * K2 + khalf * 2;
  const float* wb2 = w2t + (size_t)(32 + row) * K2 + khalf * 2;
  const float* wb3 = w2t + (size_t)(48 + row) * K2 + khalf * 2;

  v8f acc0 = {}, acc1 = {}, acc2 = {}, acc3 = {};
#pragma unroll
  for (int it = 0; it < 24; ++it) {
    const v2f a = *(const v2f*)(arow + it * 4);
    v2f bv;
    bv = *(const v2f*)(wb0 + it * 4);
    acc0 = __builtin_amdgcn_wmma_f32_16x16x4_f32(false, a, false, bv, (short)0, acc0, false, false);
    bv = *(const v2f*)(wb1 + it * 4);
    acc1 = __builtin_amdgcn_wmma_f32_16x16x4_f32(false, a, false, bv, (short)0, acc1, false, false);
    bv = *(const v2f*)(wb2 + it * 4);
    acc2 = __builtin_amdgcn_wmma_f32_16x16x4_f32(false, a, false, bv, (short)0, acc2, false, false);
    bv = *(const v2f*)(wb3 + it * 4);
    acc3 = __builtin_amdgcn_wmma_f32_16x16x4_f32(false, a, false, bv, (short)0, acc3, false, false);
  }

  const float bias0 = b2[row], bias1 = b2[16 + row], bias2 = b2[32 + row], bias3 = b2[48 + row];
  const int rbase = khalf * 8;
#pragma unroll
  for (int r = 0; r < 8; ++r) {
    const size_t o = (size_t)(mtile * 16 + rbase + r) * C2 + row;
    out[o]      = fmaxf(acc0[r] + bias0, 0.0f);
    out[o + 16] = fmaxf(acc1[r] + bias1, 0.0f);
    out[o + 32] = fmaxf(acc2[r] + bias2, 0.0f);
    out[o + 48] = fmaxf(acc3[r] + bias3, 0.0f);
  }
}

// ---------------------------------------------------------------------------
// pool3 (k=3,s=3,pad=1) + relu, with transpose to conv3's K layout.
// in:  conv2out [BN*10][64]  ([bn][p][ci])
// out: pooled   [BN][256]    ([bn][ci*4+q])
// ---------------------------------------------------------------------------
__global__ void __launch_bounds__(256) pool2_kernel(
    const float* __restrict__ in, float* __restrict__ out) {
  const int t  = blockIdx.x * 256 + threadIdx.x;   // BN*256 threads
  const int bn = t >> 8;
  const int r  = t & 255;
  const int q  = r >> 6;
  const int ci = r & 63;
  const float* base = in + (size_t)bn * (L2C * C2) + ci;
  // window q reads conv2 positions {3q-1, 3q, 3q+1}; only 3q-1 is skipped (q==0)
  float v = base[(size_t)(3 * q) * C2];
  v = fmaxf(v, base[(size_t)(3 * q + 1) * C2]);
  if (q > 0) v = fmaxf(v, base[(size_t)(3 * q - 1) * C2]);
  out[(size_t)bn * K3 + ci * 4 + q] = fmaxf(v, 0.0f);
}

// ---------------------------------------------------------------------------
// Stage C: conv3 as GEMM: M=BN, N=128, K=256, A=pooled, B=w3^T.
// feat[m][n] = relu(b3[n] + sum_k pooled[m][k]*w3[n][k])
// ---------------------------------------------------------------------------
__global__ void __launch_bounds__(128) conv3_wmma_kernel(
    const float* __restrict__ Ain, const float* __restrict__ w3,
    const float* __restrict__ b3, float* __restrict__ feat) {
  const int lane  = threadIdx.x & 31;
  const int wave  = threadIdx.x >> 5;
  const int mtile = blockIdx.x * 4 + wave;     // 2048 m-tiles
  const int row   = lane & 15;
  const int khalf = lane >> 4;
  const float* arow = Ain + (size_t)(mtile * 16 + row) * K3 + khalf * 2;

  v8f acc[8];
#pragma unroll
  for (int nt = 0; nt < 8; ++nt) acc[nt] = (v8f){};

#pragma unroll 4
  for (int it = 0; it < 64; ++it) {
    const v2f a = *(const v2f*)(arow + it * 4);
#pragma unroll
    for (int nt = 0; nt < 8; ++nt) {
      const v2f b = *(const v2f*)(w3 + (size_t)(nt * 16 + row) * K3 + khalf * 2 + it * 4);
      acc[nt] = __builtin_amdgcn_wmma_f32_16x16x4_f32(false, a, false, b, (short)0, acc[nt], false, false);
    }
  }

  const int rbase = khalf * 8;
#pragma unroll
  for (int nt = 0; nt < 8; ++nt) {
    const float bias = b3[nt * 16 + row];
#pragma unroll
    for (int r = 0; r < 8; ++r) {
      const size_t o = (size_t)(mtile * 16 + rbase + r) * C3 + nt * 16 + row;
      feat[o] = fmaxf(acc[nt][r] + bias, 0.0f);
    }
  }
}

// ---------------------------------------------------------------------------
// Head: logits = feat @ wl^T + bl ; softmax. One block per batch row.
// feat: [256][16384], wl: [3][16384]
// ---------------------------------------------------------------------------
__global__ void __launch_bounds__(256) head_kernel(
    const float* __restrict__ feat, const float* __restrict__ wl,
    const float* __restrict__ bl, float* __restrict__ out) {
  __shared__ float red0[256], red1[256], red2[256];
  const int b   = blockIdx.x;
  const int tid = threadIdx.x;
  const float* f = feat + (size_t)b * (NN * C3);
  float s0 = 0.f, s1 = 0.f, s2 = 0.f;
  for (int i = tid; i < NN * C3; i += 256) {
    const float v = f[i];
    s0 = fmaf(v, wl[i], s0);
    s1 = fmaf(v, wl[NN * C3 + i], s1);
    s2 = fmaf(v, wl[2 * NN * C3 + i], s2);
  }
  red0[tid] = s0; red1[tid] = s1; red2[tid] = s2;
  __syncthreads();
  for (int off = 128; off > 0; off >>= 1) {
    if (tid < off) {
      red0[tid] += red0[tid + off];
      red1[tid] += red1[tid + off];
      red2[tid] += red2[tid + off];
    }
    __syncthreads();
  }
  if (tid == 0) {
    const float l0 = red0[0] + bl[0];
    const float l1 = red1[0] + bl[1];
    const float l2 = red2[0] + bl[2];
    const float mx = fmaxf(l0, fmaxf(l1, l2));
    const float e0 = expf(l0 - mx), e1 = expf(l1 - mx), e2 = expf(l2 - mx);
    const float inv = 1.0f / (e0 + e1 + e2);
    out[b * 3 + 0] = e0 * inv;
    out[b * 3 + 1] = e1 * inv;
    out[b * 3 + 2] = e2 * inv;
  }
}

// ---------------------------------------------------------------------------
extern "C" void kernel_launch(void* const* d_in, const int* in_sizes, int n_in,
                              void* d_out, int out_size, void* d_ws, size_t ws_size,
                              hipStream_t stream) {
  const float* x  = (const float*)d_in[0];
  const float* w1 = (const float*)d_in[1];
  const float* b1 = (const float*)d_in[2];
  const float* w2 = (const float*)d_in[3];
  const float* b2 = (const float*)d_in[4];
  const float* w3 = (const float*)d_in[5];
  const float* b3 = (const float*)d_in[6];
  const float* wl = (const float*)d_in[7];
  const float* bl = (const float*)d_in[8];
  float* out = (float*)d_out;

  char* ws = (char*)d_ws;
  const size_t stageA_bytes   = (size_t)BN * C1 * L1 * sizeof(float);  // 125.8 MB
  const size_t conv2out_bytes = (size_t)BN * L2C * C2 * sizeof(float); //  83.9 MB
  float* stageA   = (float*)(ws);                                  // [BN*10][96]
  float* conv2out = (float*)(ws + stageA_bytes);                   // [BN*10][64]
  float* w2t      = (float*)(ws + stageA_bytes + conv2out_bytes);  // [64][96]
  float* pooled   = (float*)(ws);                                  // [BN][256] (stageA dead)
  float* feat     = (float*)(ws + (size_t)BN * K3 * sizeof(float)); // [BN][128]

  w2t_kernel<<<(C2 * K2 + 255) / 256, 256, 0, stream>>>(w2, w2t);
  conv1_pool1_kernel<<<BN, 256, 0, stream>>>(x, w1, b1, stageA);
  conv2_wmma_kernel<<<(BN * L2C / 16) / 4, 128, 0, stream>>>(stageA, w2t, b2, conv2out);
  pool2_kernel<<<BN, 256, 0, stream>>>(conv2out, pooled);
  conv3_wmma_kernel<<<(BN / 16) / 4, 128, 0, stream>>>(pooled, w3, b3, feat);
  head_kernel<<<BS, 256, 0, stream>>>(feat, wl, bl, out);
}